// GIN_84507776516707
// MI455X (gfx1250) — compile-verified
//
#include <hip/hip_runtime.h>
#include <hip/hip_bf16.h>

// ---------------------------------------------------------------------------
// GIN (2x GINConv, eps=0) on gfx1250.
//   agg = h + segment_sum(h[src], dst)
//   z   = relu(agg @ Wa + ba); h' = relu(z @ Wb + bb)
// Aggregation: float4 gather + global_atomic_add_f32 scatter (bandwidth-bound).
// GEMMs: V_WMMA_F32_16X16X4_F32; A-slab + transposed-W staged in LDS;
//        two independent accumulators per K-step to break the WMMA dep chain.
// ---------------------------------------------------------------------------

typedef __attribute__((ext_vector_type(2))) float v2f;
typedef __attribute__((ext_vector_type(8))) float v8f;

#define NFEAT 128
#define ROWS_PER_BLOCK 128   // 8 waves x 16 rows
#define GEMM_THREADS 256

// ---------------------------------------------------------------------------
// Scatter-add: 32 threads per edge, 4 floats per thread.
// ---------------------------------------------------------------------------
__global__ __launch_bounds__(256)
void gin_scatter_add(const float* __restrict__ h,
                     const int* __restrict__ src,
                     const int* __restrict__ dst,
                     float* __restrict__ agg,
                     int nEdges) {
  long gid = (long)blockIdx.x * blockDim.x + threadIdx.x;
  long e = gid >> 5;
  if (e >= nEdges) return;
  int f = (int)(gid & 31) * 4;
  long s = (long)src[e];
  long d = (long)dst[e];
  const float4 v = *(const float4*)(h + s * NFEAT + f);
  float* base = agg + d * NFEAT + f;
  atomicAdd(base + 0, v.x);
  atomicAdd(base + 1, v.y);
  atomicAdd(base + 2, v.z);
  atomicAdd(base + 3, v.w);
}

// ---------------------------------------------------------------------------
// Y[M,128] = relu(X[M,128] @ W[128,128] + bias)
// Block = 256 threads = 8 waves; each block owns a 128-row slab.
// LDS: sA  = 128x128 f32 slab of X, row major            (64 KB)
//      sWT = W transposed (sWT[n*128+k] = W[k*128+n])    (64 KB)
//      sB  = bias                                        (0.5 KB)
// Wave w computes rows [w*16, w*16+16); N-tiles processed in pairs with two
// independent accumulators so consecutive WMMAs are not RAW-dependent.
//
// f32 WMMA operand layouts (ISA 7.12.2):
//   A 16x4 : lane<16 -> row=lane,    K = {4k+0, 4k+1}
//            lane>=16-> row=lane-16, K = {4k+2, 4k+3}
//   B 4x16 : lane<16 -> col=lane,    K = {4k+0, 4k+1}
//            lane>=16-> col=lane-16, K = {4k+2, 4k+3}
//   C/D    : vgpr v, lane<16 -> (M=v,   N=lane);  lane>=16 -> (M=v+8, N=lane-16)
// ---------------------------------------------------------------------------
__global__ __launch_bounds__(GEMM_THREADS)
void gin_gemm_bias_relu(const float* __restrict__ X,
                        const float* __restrict__ W,
                        const float* __restrict__ bias,
                        float* __restrict__ Y,
                        int M) {
  __shared__ float sA[ROWS_PER_BLOCK * NFEAT];
  __shared__ float sWT[NFEAT * NFEAT];
  __shared__ float sB[NFEAT];

  const int tid  = threadIdx.x;
  const int wave = tid >> 5;
  const int lane = tid & 31;
  const long rowBase = (long)blockIdx.x * ROWS_PER_BLOCK;

  // Stage W transposed into LDS.
  for (int i = tid; i < NFEAT * NFEAT; i += GEMM_THREADS) {
    int k = i >> 7;
    int n = i & (NFEAT - 1);
    sWT[n * NFEAT + k] = W[i];
  }
  if (tid < NFEAT) sB[tid] = bias[tid];

  // Stage the 128-row X slab (zero-pad past M).
  for (int i = tid; i < ROWS_PER_BLOCK * NFEAT; i += GEMM_THREADS) {
    long gr = rowBase + (i >> 7);
    sA[i] = (gr < M) ? X[gr * NFEAT + (i & (NFEAT - 1))] : 0.0f;
  }
  __syncthreads();

  const int r0    = wave * 16;          // wave's row tile inside slab
  const int l16   = lane & 15;
  const int halfK = (lane >> 4) << 1;   // 0 for lanes 0-15, 2 for lanes 16-31
  const int mOff  = (lane >> 4) << 3;   // C-layout row offset

  const float* aRow = &sA[(r0 + l16) * NFEAT + halfK];

  // Process N-tiles in pairs: two independent accumulators per K-step.
  for (int n = 0; n < 8; n += 2) {
    const int col0 = n * 16 + l16;
    const int col1 = col0 + 16;
    const float* bCol0 = &sWT[col0 * NFEAT + halfK];
    const float* bCol1 = &sWT[col1 * NFEAT + halfK];

    v8f acc0 = {};
    v8f acc1 = {};
#pragma unroll
    for (int kk = 0; kk < 32; ++kk) {
      v2f a  = *(const v2f*)(aRow  + kk * 4);   // ds_load_b64 (8B aligned)
      v2f b0 = *(const v2f*)(bCol0 + kk * 4);   // ds_load_b64
      v2f b1 = *(const v2f*)(bCol1 + kk * 4);   // ds_load_b64
      acc0 = __builtin_amdgcn_wmma_f32_16x16x4_f32(
          false, a, false, b0, (short)0, acc0, false, false);
      acc1 = __builtin_amdgcn_wmma_f32_16x16x4_f32(
          false, a, false, b1, (short)0, acc1, false, false);
    }

    const float bv0 = sB[col0];
    const float bv1 = sB[col1];
#pragma unroll
    for (int v = 0; v < 8; ++v) {
      long gr = rowBase + r0 + mOff + v;
      if (gr < M) {
        float val0 = acc0[v] + bv0;
        float val1 = acc1[v] + bv1;
        Y[gr * NFEAT + col0] = val0 > 0.0f ? val0 : 0.0f;
        Y[gr * NFEAT + col1] = val1 > 0.0f ? val1 : 0.0f;
      }
    }
  }
}

// ---------------------------------------------------------------------------
// Launch: Z (ws) and d_out ping-pong as feature buffers.
// ---------------------------------------------------------------------------
extern "C" void kernel_launch(void* const* d_in, const int* in_sizes, int n_in,
                              void* d_out, int out_size, void* d_ws, size_t ws_size,
                              hipStream_t stream) {
  const float* x   = (const float*)d_in[0];
  const int*   ei  = (const int*)d_in[1];   // [2, E]: row 0 = src, row 1 = dst
  const float* W1a = (const float*)d_in[2];
  const float* b1a = (const float*)d_in[3];
  const float* W1b = (const float*)d_in[4];
  const float* b1b = (const float*)d_in[5];
  const float* W2a = (const float*)d_in[6];
  const float* b2a = (const float*)d_in[7];
  const float* W2b = (const float*)d_in[8];
  const float* b2b = (const float*)d_in[9];

  const int nNodes = in_sizes[0] / NFEAT;       // 50000
  const int nEdges = in_sizes[1] / 2;           // 800000
  const int* src = ei;
  const int* dst = ei + nEdges;

  float* Z = (float*)d_ws;                      // 50000*128*4 = 25.6 MB
  float* O = (float*)d_out;
  const size_t featBytes = (size_t)nNodes * NFEAT * sizeof(float);

  const int scatterBlocks = (int)(((long)nEdges * 32 + 255) / 256);
  const int gemmBlocks    = (nNodes + ROWS_PER_BLOCK - 1) / ROWS_PER_BLOCK;

  // ---- Layer 1 ----
  hipMemcpyAsync(Z, x, featBytes, hipMemcpyDeviceToDevice, stream);   // Z = x (self)
  gin_scatter_add<<<scatterBlocks, 256, 0, stream>>>(x, src, dst, Z, nEdges);
  gin_gemm_bias_relu<<<gemmBlocks, GEMM_THREADS, 0, stream>>>(Z, W1a, b1a, O, nNodes);
  gin_gemm_bias_relu<<<gemmBlocks, GEMM_THREADS, 0, stream>>>(O, W1b, b1b, Z, nNodes);
  // (post-conv ReLU is a no-op after the MLP's final ReLU)

  // ---- Layer 2 ----  (Z holds h1)
  hipMemcpyAsync(O, Z, featBytes, hipMemcpyDeviceToDevice, stream);   // O = h1 (self)
  gin_scatter_add<<<scatterBlocks, 256, 0, stream>>>(Z, src, dst, O, nEdges);
  gin_gemm_bias_relu<<<gemmBlocks, GEMM_THREADS, 0, stream>>>(O, W2a, b2a, Z, nNodes);
  gin_gemm_bias_relu<<<gemmBlocks, GEMM_THREADS, 0, stream>>>(Z, W2b, b2b, O, nNodes);
}